// Diagnosis_1640677507712
// MI455X (gfx1250) — compile-verified
//
#include <hip/hip_runtime.h>
#include <hip/hip_bf16.h>
#include <math.h>

typedef __attribute__((ext_vector_type(2))) float v2f;
typedef __attribute__((ext_vector_type(8))) float v8f;

#define CDIM 128

// ---------------------------------------------------------------------------
// GEMM: P[M,128] = X[M,128] @ W[128,128]^T   (P[m,n] = sum_k X[m,k]*W[n,k])
// One wave computes a 16x128 strip using V_WMMA_F32_16X16X4_F32.
// f32 A 16x4 fragment: lanes0-15 hold K={k,k+1}, lanes16-31 K={k+2,k+3}.
// B = W^T 4x16 fragment is symmetric: float2 at W[n*128 + k + 2*half].
// ---------------------------------------------------------------------------
__global__ __launch_bounds__(256) void gemm_xwT_wmma(
    const float* __restrict__ X, const float* __restrict__ W,
    float* __restrict__ P, int Mrows) {
  const int wave = (blockIdx.x * blockDim.x + threadIdx.x) >> 5;
  const int lane = threadIdx.x & 31;
  const int li   = lane & 15;
  const int half = lane >> 4;
  const int tiles = (Mrows + 15) >> 4;
  if (wave >= tiles) return;

  int m = wave * 16 + li;
  if (m >= Mrows) m = Mrows - 1;          // clamp for tail tile (loads only)
  const float* __restrict__ xrow = X + (size_t)m * CDIM;

  const float* wrow[8];
#pragma unroll
  for (int t = 0; t < 8; ++t)
    wrow[t] = W + (size_t)(t * 16 + li) * CDIM;

  v8f acc[8] = {};
  for (int k = 0; k < CDIM; k += 4) {
    v2f a = *(const v2f*)(xrow + k + 2 * half);
#pragma unroll
    for (int t = 0; t < 8; ++t) {
      v2f b = *(const v2f*)(wrow[t] + k + 2 * half);
      acc[t] = __builtin_amdgcn_wmma_f32_16x16x4_f32(
          /*neg_a=*/false, a, /*neg_b=*/false, b,
          /*c_mod=*/(short)0, acc[t], /*reuse_a=*/false, /*reuse_b=*/false);
    }
  }

  // C/D layout: c[g] holds row (g + 8*half), col = li (per 16-lane half)
#pragma unroll
  for (int t = 0; t < 8; ++t) {
#pragma unroll
    for (int g = 0; g < 8; ++g) {
      int r = wave * 16 + g + 8 * half;
      if (r < Mrows) P[(size_t)r * CDIM + t * 16 + li] = acc[t][g];
    }
  }
}

// ---------------------------------------------------------------------------
// Zero a segment accumulator (seg_sum or seg_cnt, each zeroed by its own launch).
// ---------------------------------------------------------------------------
__global__ void zero_kernel(float* __restrict__ p, int n) {
  int i = blockIdx.x * blockDim.x + threadIdx.x;
  if (i < n) p[i] = 0.0f;
}

__device__ __forceinline__ float sigmoidf(float x) {
  return 1.0f / (1.0f + __expf(-x));
}

// ---------------------------------------------------------------------------
// Edge phase: one wave per edge. Lane l covers channels [4l,4l+4).
// scalar_e = dot(sigmoid(cs+sp+b_stu) - sigmoid(ci+ip+b_item), w_pred)
// Atomically accumulate scalar_e and 1.0 into segment buckets.
// ---------------------------------------------------------------------------
__global__ __launch_bounds__(256) void edge_kernel(
    const float* __restrict__ conc_s, const float* __restrict__ stu_p,
    const float* __restrict__ conc_i, const float* __restrict__ item_p,
    const float* __restrict__ b_stu, const float* __restrict__ b_item,
    const float* __restrict__ w_pred,
    const int* __restrict__ stu_track, const int* __restrict__ item_index,
    const int* __restrict__ conc_index, const int* __restrict__ mean_index,
    float* __restrict__ seg_sum, float* __restrict__ seg_cnt, int E) {
  const int lane = threadIdx.x & 31;
  const int wid  = (blockIdx.x * blockDim.x + threadIdx.x) >> 5;
  const int nwav = (gridDim.x * blockDim.x) >> 5;

  const float4 bs = ((const float4*)b_stu)[lane];
  const float4 bi = ((const float4*)b_item)[lane];
  const float4 w  = ((const float4*)w_pred)[lane];

  for (int e = wid; e < E; e += nwav) {
    const int s  = stu_track[e];
    const int it = item_index[e];
    const int cn = conc_index[e];
    const int g  = mean_index[e];

    const float4 cs = ((const float4*)(conc_s + (size_t)cn * CDIM))[lane];
    const float4 sp = ((const float4*)(stu_p  + (size_t)s  * CDIM))[lane];
    const float4 ci = ((const float4*)(conc_i + (size_t)cn * CDIM))[lane];
    const float4 ip = ((const float4*)(item_p + (size_t)it * CDIM))[lane];

    float4 d;
    d.x = sigmoidf(cs.x + sp.x + bs.x) - sigmoidf(ci.x + ip.x + bi.x);
    d.y = sigmoidf(cs.y + sp.y + bs.y) - sigmoidf(ci.y + ip.y + bi.y);
    d.z = sigmoidf(cs.z + sp.z + bs.z) - sigmoidf(ci.z + ip.z + bi.z);
    d.w = sigmoidf(cs.w + sp.w + bs.w) - sigmoidf(ci.w + ip.w + bi.w);

    float part = d.x * w.x + d.y * w.y + d.z * w.z + d.w * w.w;
#pragma unroll
    for (int off = 16; off > 0; off >>= 1)
      part += __shfl_down(part, off, 32);

    if (lane == 0) {
      atomicAdd(&seg_sum[g], part);
      atomicAdd(&seg_cnt[g], 1.0f);
    }
  }
}

// ---------------------------------------------------------------------------
// Final: out[m] = sigmoid(seg_sum[m]/max(cnt,1) + b_pred)
// ---------------------------------------------------------------------------
__global__ void final_kernel(const float* __restrict__ seg_sum,
                             const float* __restrict__ seg_cnt,
                             const float* __restrict__ b_pred,
                             float* __restrict__ out, int M) {
  int m = blockIdx.x * blockDim.x + threadIdx.x;
  if (m < M) {
    float c = seg_cnt[m];
    float v = seg_sum[m] / fmaxf(c, 1.0f) + b_pred[0];
    out[m] = sigmoidf(v);
  }
}

// ---------------------------------------------------------------------------
extern "C" void kernel_launch(void* const* d_in, const int* in_sizes, int n_in,
                              void* d_out, int out_size, void* d_ws, size_t ws_size,
                              hipStream_t stream) {
  const float* stu_x    = (const float*)d_in[0];
  const float* item_x   = (const float*)d_in[1];
  const float* conc_x   = (const float*)d_in[2];
  const float* W_stu    = (const float*)d_in[3];
  const float* b_stu    = (const float*)d_in[4];
  const float* W_item   = (const float*)d_in[5];
  const float* b_item   = (const float*)d_in[6];
  const float* W_pred   = (const float*)d_in[7];
  const float* b_pred   = (const float*)d_in[8];
  const int*   stu_track  = (const int*)d_in[9];
  const int*   item_index = (const int*)d_in[10];
  const int*   conc_index = (const int*)d_in[11];
  const int*   mean_index = (const int*)d_in[12];

  const int NS = in_sizes[0] / CDIM;
  const int NI = in_sizes[1] / CDIM;
  const int NC = in_sizes[2] / CDIM;
  const int E  = in_sizes[9];
  const int M  = out_size;

  // Workspace layout (256B aligned slabs)
  auto align256 = [](size_t x) { return (x + 255) & ~(size_t)255; };
  char* ws = (char*)d_ws;
  size_t off = 0;
  float* stu_p  = (float*)(ws + off); off = align256(off + (size_t)NS * CDIM * 4);
  float* item_p = (float*)(ws + off); off = align256(off + (size_t)NI * CDIM * 4);
  float* conc_s = (float*)(ws + off); off = align256(off + (size_t)NC * CDIM * 4);
  float* conc_i = (float*)(ws + off); off = align256(off + (size_t)NC * CDIM * 4);
  float* seg_sum = (float*)(ws + off); off = align256(off + (size_t)M * 4);
  float* seg_cnt = (float*)(ws + off); off = align256(off + (size_t)M * 4);

  // 1) zero segment accumulators (sum + cnt)
  {
    int n = 2 * M;  // seg_sum and seg_cnt are adjacent only logically; zero both
    zero_kernel<<<(M + 255) / 256, 256, 0, stream>>>(seg_sum, M);
    zero_kernel<<<(M + 255) / 256, 256, 0, stream>>>(seg_cnt, M);
    (void)n;
  }

  // 2) projections (WMMA f32 GEMMs): P = X @ W^T
  auto launch_gemm = [&](const float* X, const float* W, float* P, int Mrows) {
    int tiles = (Mrows + 15) / 16;           // one wave per 16-row strip
    int blocks = (tiles + 7) / 8;            // 8 waves (256 threads) per block
    gemm_xwT_wmma<<<blocks, 256, 0, stream>>>(X, W, P, Mrows);
  };
  launch_gemm(stu_x,  W_stu,  stu_p,  NS);
  launch_gemm(conc_x, W_stu,  conc_s, NC);
  launch_gemm(item_x, W_item, item_p, NI);
  launch_gemm(conc_x, W_item, conc_i, NC);

  // 3) fused edge phase (gather + sigmoid + dot + segment scatter)
  {
    int waves = E;                           // one wave per edge
    int blocks = (waves + 7) / 8;
    edge_kernel<<<blocks, 256, 0, stream>>>(conc_s, stu_p, conc_i, item_p,
                                            b_stu, b_item, W_pred,
                                            stu_track, item_index, conc_index,
                                            mean_index, seg_sum, seg_cnt, E);
  }

  // 4) final mean + bias + sigmoid
  final_kernel<<<(M + 255) / 256, 256, 0, stream>>>(seg_sum, seg_cnt, b_pred,
                                                    (float*)d_out, M);
}